// InstanceFusion_18047452578379
// MI455X (gfx1250) — compile-verified
//
#include <hip/hip_runtime.h>
#include <hip/hip_bf16.h>
#include <math.h>

// ---- problem constants (match reference) ----
#define HEADS 2
#define DHEAD 32
#define HD    64     // HEADS*DHEAD
#define ED2   128    // HEADS * 2*DHEAD (ft2 row)

typedef float v2f __attribute__((ext_vector_type(2)));
typedef float v8f __attribute__((ext_vector_type(8)));

// ---------------------------------------------------------------------------
// WMMA f32 GEMM:  C[M x 64] = A[M x K] @ W[K x 64] + bias[64]
// One block = 128 threads = 4 waves; each wave owns one 16-col tile, block
// owns 16 rows. All fragments preloaded, then a clean back-to-back chain of
// V_WMMA_F32_16X16X4_F32. Out-of-range rows are CLAMPED (not masked) so the
// whole kernel is convergent; garbage rows are simply never stored.
// ---------------------------------------------------------------------------
template <int K>
__global__ __launch_bounds__(128)
void gemm_n64_wmma(const float* __restrict__ A, const float* __restrict__ W,
                   const float* __restrict__ bias, float* __restrict__ C, int M) {
    __shared__ float sW[K * 64];
    __shared__ float sB[64];
    for (int i = threadIdx.x * 4; i < K * 64; i += 128 * 4)
        *(float4*)&sW[i] = *(const float4*)&W[i];
    if (threadIdx.x < 16)
        ((float4*)sB)[threadIdx.x] = ((const float4*)bias)[threadIdx.x];
    __syncthreads();

    const int wave = threadIdx.x >> 5;   // 0..3 -> 16-col tile
    const int lane = threadIdx.x & 31;
    const int lrow = lane & 15;          // M index within tile (A) / col (B,C)
    const int hi   = lane >> 4;          // 0: K=k,k+1   1: K=k+2,k+3
    const int row0 = blockIdx.x * 16;
    const int col0 = wave * 16;

    int arow = row0 + lrow;
    if (arow >= M) arow = M - 1;                 // clamp: convergent, never stored
    const float* Arow = A + (size_t)arow * K;

    constexpr int NK = K / 4;
    v2f afrag[NK], bfrag[NK];
    #pragma unroll
    for (int i = 0; i < NK; ++i) {
        // A fragment (16x4 f32): lanes 0-15 -> K=4i,4i+1 ; lanes 16-31 -> 4i+2,4i+3
        afrag[i] = *(const v2f*)&Arow[4 * i + 2 * hi];
        // B fragment (4x16 f32): mirrored K split, row striped across lanes
        bfrag[i].x = sW[(4 * i + 2 * hi + 0) * 64 + col0 + lrow];
        bfrag[i].y = sW[(4 * i + 2 * hi + 1) * 64 + col0 + lrow];
    }

    v8f acc = {};
    #pragma unroll
    for (int i = 0; i < NK; ++i)
        acc = __builtin_amdgcn_wmma_f32_16x16x4_f32(
                  false, afrag[i], false, bfrag[i], (short)0, acc, false, false);

    // C/D layout: VGPR v holds row = v (+8 for lanes 16-31), col = lane%16
    const int crow = row0 + 8 * hi;
    const int ccol = col0 + lrow;
    const float bb = sB[ccol];
    #pragma unroll
    for (int v = 0; v < 8; ++v) {
        int r = crow + v;
        if (r < M) C[(size_t)r * 64 + ccol] = acc[v] + bb;
    }
}

// ---------------------------------------------------------------------------
__global__ void fill_f32(float* __restrict__ p, float v, size_t n) {
    size_t i = (size_t)blockIdx.x * blockDim.x + threadIdx.x;
    if (i < n) p[i] = v;
}

// per-node GAT logits: e_src/e_dst = sum_d attn * relu(hp)
__global__ void node_logits(const float* __restrict__ hp,
                            const float* __restrict__ attn_src,
                            const float* __restrict__ attn_dst,
                            float* __restrict__ e_src, float* __restrict__ e_dst, int n) {
    int i = blockIdx.x * blockDim.x + threadIdx.x;
    if (i >= n) return;
    const float4* row = (const float4*)(hp + (size_t)i * HD);
    const float4* as4 = (const float4*)attn_src;
    const float4* ad4 = (const float4*)attn_dst;
    #pragma unroll
    for (int h = 0; h < HEADS; ++h) {
        float s = 0.f, d = 0.f;
        #pragma unroll
        for (int q = 0; q < DHEAD / 4; ++q) {
            float4 v = row[h * 8 + q];
            float4 a = as4[h * 8 + q];
            float4 b = ad4[h * 8 + q];
            float vx = fmaxf(v.x, 0.f), vy = fmaxf(v.y, 0.f);
            float vz = fmaxf(v.z, 0.f), vw = fmaxf(v.w, 0.f);
            s += a.x * vx + a.y * vy + a.z * vz + a.w * vw;
            d += b.x * vx + b.y * vy + b.z * vz + b.w * vw;
        }
        e_src[(size_t)i * HEADS + h] = s;
        e_dst[(size_t)i * HEADS + h] = d;
    }
}

// sign-aware float atomic max (works for mixed-sign values)
__device__ inline void atomicMaxF(float* addr, float value) {
    if (value >= 0.0f) {
        atomicMax((int*)addr, __float_as_int(value));
    } else {
        atomicMin((unsigned int*)addr, __float_as_uint(value));
    }
}

// pass 1: per-edge logit (u_add_v + edge-feature term), running segment max
__global__ void edge_logits(const float* __restrict__ e_src, const float* __restrict__ e_dst,
                            const float* __restrict__ mid_f, const float* __restrict__ rel_f,
                            const float* __restrict__ attn_edge,
                            const int* __restrict__ src, const int* __restrict__ dst,
                            const int* __restrict__ hn, const int* __restrict__ he,
                            float* __restrict__ elog, float* __restrict__ m, int E_) {
    int e = blockIdx.x * blockDim.x + threadIdx.x;
    if (e >= E_) return;
    const int s = src[e], d = dst[e], nn = hn[e], rr = he[e];
    const float4* mf = (const float4*)(mid_f + (size_t)nn * HD);
    const float4* rf = (const float4*)(rel_f + (size_t)rr * HD);
    const float4* ae = (const float4*)attn_edge;   // [H][2*DHEAD]
    #pragma unroll
    for (int h = 0; h < HEADS; ++h) {
        float acc = e_src[(size_t)s * HEADS + h] + e_dst[(size_t)d * HEADS + h];
        #pragma unroll
        for (int q = 0; q < DHEAD / 4; ++q) {
            float4 v1 = mf[h * 8 + q];
            float4 v2 = rf[h * 8 + q];
            float4 a1 = ae[h * 16 + q];
            float4 a2 = ae[h * 16 + 8 + q];
            acc += a1.x * fmaxf(v1.x, 0.f) + a1.y * fmaxf(v1.y, 0.f)
                 + a1.z * fmaxf(v1.z, 0.f) + a1.w * fmaxf(v1.w, 0.f);
            acc += a2.x * fmaxf(v2.x, 0.f) + a2.y * fmaxf(v2.y, 0.f)
                 + a2.z * fmaxf(v2.z, 0.f) + a2.w * fmaxf(v2.w, 0.f);
        }
        elog[(size_t)e * HEADS + h] = acc;
        atomicMaxF(&m[(size_t)d * HEADS + h], acc);
    }
}

// pass 2: exp(e - max), running segment sum (denominator)
__global__ void edge_exp(float* __restrict__ elog, const float* __restrict__ m,
                         const int* __restrict__ dst, float* __restrict__ denom, int E_) {
    int e = blockIdx.x * blockDim.x + threadIdx.x;
    if (e >= E_) return;
    const int d = dst[e];
    #pragma unroll
    for (int h = 0; h < HEADS; ++h) {
        float ex = __expf(elog[(size_t)e * HEADS + h] - m[(size_t)d * HEADS + h]);
        elog[(size_t)e * HEADS + h] = ex;
        atomicAdd(&denom[(size_t)d * HEADS + h], ex);
    }
}

// pass 3: normalized attention scatter-add of hp[src] and h_e into rst / ft2.
// One lane per (edge, d); atomics resolve in L2 (everything L2-resident).
__global__ void aggregate(const float* __restrict__ ex, const float* __restrict__ denom,
                          const float* __restrict__ hp, const float* __restrict__ mid_f,
                          const float* __restrict__ rel_f,
                          const int* __restrict__ src, const int* __restrict__ dst,
                          const int* __restrict__ hn, const int* __restrict__ he,
                          float* __restrict__ rst, float* __restrict__ ft2, int E_) {
    size_t t = (size_t)blockIdx.x * blockDim.x + threadIdx.x;
    int e = (int)(t >> 5);
    int d = (int)(t & 31);
    if (e >= E_) return;
    const int s = src[e], dn = dst[e], nn = hn[e], rr = he[e];
    #pragma unroll
    for (int h = 0; h < HEADS; ++h) {
        float a = ex[(size_t)e * HEADS + h] / denom[(size_t)dn * HEADS + h];
        atomicAdd(&rst[(size_t)dn * HD + h * DHEAD + d],
                  hp[(size_t)s * HD + h * DHEAD + d] * a);
        atomicAdd(&ft2[(size_t)dn * ED2 + h * 2 * DHEAD + d],
                  mid_f[(size_t)nn * HD + h * DHEAD + d] * a);
        atomicAdd(&ft2[(size_t)dn * ED2 + h * 2 * DHEAD + DHEAD + d],
                  rel_f[(size_t)rr * HD + h * DHEAD + d] * a);
    }
}

// pass 4: rst += ft2 @ W_edge + b_edge ; relu(rst + hp) ; row L2 normalize.
// One block (64 threads = 2 waves) per node; ft2 row + W_edge staged in LDS.
__global__ __launch_bounds__(64)
void finalize(const float* __restrict__ rst, const float* __restrict__ ft2,
              const float* __restrict__ hp, const float* __restrict__ W_edge,
              const float* __restrict__ b_edge, float* __restrict__ out, int n) {
    __shared__ float sWe[64 * DHEAD];
    __shared__ float sf2[ED2];
    __shared__ float red[64];
    const int node = blockIdx.x;
    const int t = threadIdx.x;            // 0..63 == h*32 + c
    if (node >= n) return;
    #pragma unroll
    for (int i = 0; i < 8; ++i)           // 64 threads * 8 float4 = 2048 floats
        ((float4*)sWe)[t + i * 64] = ((const float4*)W_edge)[t + i * 64];
    sf2[t]      = ft2[(size_t)node * ED2 + t];
    sf2[64 + t] = ft2[(size_t)node * ED2 + 64 + t];
    __syncthreads();
    const int h = t >> 5;
    const int c = t & 31;
    float acc = rst[(size_t)node * HD + t] + b_edge[c] + hp[(size_t)node * HD + t];
    #pragma unroll
    for (int k = 0; k < 64; ++k)
        acc += sf2[h * 64 + k] * sWe[k * DHEAD + c];
    float o = fmaxf(acc, 0.0f);
    red[t] = o * o;
    __syncthreads();
    for (int sft = 32; sft > 0; sft >>= 1) {
        if (t < sft) red[t] += red[t + sft];
        __syncthreads();
    }
    float inv = 1.0f / fmaxf(sqrtf(red[0]), 1e-12f);
    out[(size_t)node * HD + t] = o * inv;
}

// ---------------------------------------------------------------------------
extern "C" void kernel_launch(void* const* d_in, const int* in_sizes, int n_in,
                              void* d_out, int out_size, void* d_ws, size_t ws_size,
                              hipStream_t stream) {
    const float* h_in     = (const float*)d_in[0];
    const float* W_t      = (const float*)d_in[1];
    const float* b_t      = (const float*)d_in[2];
    const float* nfeat    = (const float*)d_in[3];
    const float* W_mid    = (const float*)d_in[4];
    const float* b_mid    = (const float*)d_in[5];
    const float* efeat    = (const float*)d_in[6];
    const float* W_rel    = (const float*)d_in[7];
    const float* b_rel    = (const float*)d_in[8];
    const float* attn_src = (const float*)d_in[9];
    const float* attn_dst = (const float*)d_in[10];
    const float* attn_edge= (const float*)d_in[11];
    const float* W_edge   = (const float*)d_in[12];
    const float* b_edge   = (const float*)d_in[13];
    const int*   src_idx  = (const int*)d_in[14];
    const int*   dst_idx  = (const int*)d_in[15];
    const int*   hn       = (const int*)d_in[16];
    const int*   he       = (const int*)d_in[17];
    float* out = (float*)d_out;

    const int n    = in_sizes[0] / 64;   // N = 50000
    const int erel = in_sizes[6] / 32;   // EREL = 200000
    const int E_   = in_sizes[14];       // E = 800000

    // workspace carve-up (floats)
    float* ws   = (float*)d_ws;
    float* hp   = ws;                         // N*64
    float* midf = hp   + (size_t)n * HD;      // N*64
    float* relf = midf + (size_t)n * HD;      // EREL*64
    float* esrc = relf + (size_t)erel * HD;   // N*2
    float* edst = esrc + (size_t)n * HEADS;   // N*2
    float* mbuf = edst + (size_t)n * HEADS;   // N*2
    float* den  = mbuf + (size_t)n * HEADS;   // N*2
    float* elog = den  + (size_t)n * HEADS;   // E*2 (logits, then softmax numerators)
    float* rstb = elog + (size_t)E_ * HEADS;  // N*64
    float* ft2b = rstb + (size_t)n * HD;      // N*128 (contiguous after rstb)

    // projections (WMMA f32)
    gemm_n64_wmma<64><<<(n + 15) / 16, 128, 0, stream>>>(h_in, W_t, b_t, hp, n);
    gemm_n64_wmma<64><<<(n + 15) / 16, 128, 0, stream>>>(nfeat, W_mid, b_mid, midf, n);
    gemm_n64_wmma<32><<<(erel + 15) / 16, 128, 0, stream>>>(efeat, W_rel, b_rel, relf, erel);

    // per-node logits
    node_logits<<<(n + 255) / 256, 256, 0, stream>>>(hp, attn_src, attn_dst, esrc, edst, n);

    // accumulator init (every call: harness does not re-zero scratch)
    {
        size_t nm = (size_t)n * HEADS;
        fill_f32<<<(unsigned)((nm + 255) / 256), 256, 0, stream>>>(mbuf, -1e30f, nm);
        fill_f32<<<(unsigned)((nm + 255) / 256), 256, 0, stream>>>(den, 0.0f, nm);
        size_t nz = (size_t)n * (HD + ED2);   // rst + ft2 contiguous
        fill_f32<<<(unsigned)((nz + 255) / 256), 256, 0, stream>>>(rstb, 0.0f, nz);
    }

    // edge softmax + aggregation
    edge_logits<<<(E_ + 255) / 256, 256, 0, stream>>>(esrc, edst, midf, relf, attn_edge,
                                                      src_idx, dst_idx, hn, he,
                                                      elog, mbuf, E_);
    edge_exp<<<(E_ + 255) / 256, 256, 0, stream>>>(elog, mbuf, dst_idx, den, E_);
    {
        size_t tot = (size_t)E_ * 32;
        aggregate<<<(unsigned)((tot + 255) / 256), 256, 0, stream>>>(
            elog, den, hp, midf, relf, src_idx, dst_idx, hn, he, rstb, ft2b, E_);
    }

    // fc_edge + residual + relu + l2 norm
    finalize<<<n, 64, 0, stream>>>(rstb, ft2b, hp, W_edge, b_edge, out, n);
}